// GraphEncoder_old_29781303230610
// MI455X (gfx1250) — compile-verified
//
#include <hip/hip_runtime.h>
#include <hip/hip_bf16.h>

// ---------------- problem constants ----------------
#define N_NODES 50000
#define N_EDGES 1600000
#define D 64            // D_IN == D_OUT == 64
#define N_REL 6
#define N_STRIPS (N_NODES / 16)          // 3125 exact
#define WAVES_PER_BLOCK 8
#define GEMM_BLOCKS ((N_STRIPS + WAVES_PER_BLOCK - 1) / WAVES_PER_BLOCK)  // 391
#define WT_STRIDE 66                     // padded LDS row stride (floats)

typedef float v2f __attribute__((ext_vector_type(2)));
typedef float v8f __attribute__((ext_vector_type(8)));

// ---------------- phase 1: zero (dst,rel) counters ----------------
__global__ void rgcn_zero_counts(int* __restrict__ counts) {
    int i = blockIdx.x * blockDim.x + threadIdx.x;
    if (i < N_NODES * N_REL) counts[i] = 0;
}

// ---------------- phase 2: count in-edges per (dst, rel) ----------------
__global__ void rgcn_count_edges(const int* __restrict__ dstArr,
                                 const int* __restrict__ etype,
                                 int* __restrict__ counts) {
    int e = blockIdx.x * blockDim.x + threadIdx.x;
    if (e >= N_EDGES) return;
    atomicAdd(&counts[dstArr[e] * N_REL + etype[e]], 1);
}

// ---------------- phase 3: WMMA GEMMs ----------------
// gridDim.y = 7: y in [0,5] -> xr[y] = x @ W[y]  (to workspace)
//                y == 6     -> out   = x @ root + bias
// Each wave computes a 16-row x 64-col strip with V_WMMA_F32_16X16X4_F32.
__global__ void __launch_bounds__(256)
rgcn_gemm_wmma(const float* __restrict__ x,     // [N_NODES][64]
               const float* __restrict__ W,     // [6][64][64]
               const float* __restrict__ root,  // [64][64]
               const float* __restrict__ bias,  // [64]
               float* __restrict__ out,         // [N_NODES][64]
               float* __restrict__ xr) {        // [6][N_NODES][64]
    const int rel = blockIdx.y;
    const bool isRoot = (rel == N_REL);
    const float* __restrict__ B = isRoot ? root : (W + (size_t)rel * D * D);

    // Stage W transposed (Wt[n][k]) in LDS with padded stride -> B fragment
    // becomes one aligned 8-byte read per lane.
    __shared__ float Wt[D * WT_STRIDE];
    for (int i = threadIdx.x; i < D * D; i += blockDim.x) {
        int k = i >> 6, n = i & 63;
        Wt[n * WT_STRIDE + k] = B[i];
    }
    __syncthreads();

    const int wave = threadIdx.x >> 5;
    const int lane = threadIdx.x & 31;
    const int strip = blockIdx.x * WAVES_PER_BLOCK + wave;   // wave-uniform
    if (strip >= N_STRIPS) return;                           // whole-wave exit
    const int rowBase = strip * 16;

    const int half = lane >> 4;      // 0: K={k0,k0+1}, 1: K={k0+2,k0+3}
    const int l16  = lane & 15;      // row (A) / column-in-tile (B, C/D)
    const int koff = 2 * half;

    // Accumulators: four 16x16 tiles covering N = 0..63
    v8f acc0 = {}, acc1 = {}, acc2 = {}, acc3 = {};
    if (isRoot) {
        const float b0 = bias[l16], b1 = bias[16 + l16],
                    b2 = bias[32 + l16], b3 = bias[48 + l16];
#pragma unroll
        for (int v = 0; v < 8; ++v) {
            acc0[v] = b0; acc1[v] = b1; acc2[v] = b2; acc3[v] = b3;
        }
    }

    // A-matrix 16x4 f32 layout: lanes 0-15 row M=l16 hold K={k0,k0+1},
    // lanes 16-31 same rows hold K={k0+2,k0+3}.
    const float* __restrict__ aRow = x + (size_t)(rowBase + l16) * D + koff;

#pragma unroll
    for (int k0 = 0; k0 < D; k0 += 4) {
        const v2f a = *(const v2f*)(aRow + k0);
        const v2f b0 = *(const v2f*)(&Wt[(      l16) * WT_STRIDE + k0 + koff]);
        const v2f b1 = *(const v2f*)(&Wt[(16 + l16) * WT_STRIDE + k0 + koff]);
        const v2f b2 = *(const v2f*)(&Wt[(32 + l16) * WT_STRIDE + k0 + koff]);
        const v2f b3 = *(const v2f*)(&Wt[(48 + l16) * WT_STRIDE + k0 + koff]);
        acc0 = __builtin_amdgcn_wmma_f32_16x16x4_f32(false, a, false, b0, (short)0, acc0, false, false);
        acc1 = __builtin_amdgcn_wmma_f32_16x16x4_f32(false, a, false, b1, (short)0, acc1, false, false);
        acc2 = __builtin_amdgcn_wmma_f32_16x16x4_f32(false, a, false, b2, (short)0, acc2, false, false);
        acc3 = __builtin_amdgcn_wmma_f32_16x16x4_f32(false, a, false, b3, (short)0, acc3, false, false);
    }

    // C/D layout: VGPR v, lanes 0-15 -> (M=v, N=l16); lanes 16-31 -> (M=8+v, N=l16)
    float* __restrict__ dst =
        isRoot ? out : (xr + (size_t)rel * N_NODES * D);
    const int mBase = rowBase + (half ? 8 : 0);
#pragma unroll
    for (int v = 0; v < 8; ++v) {
        float* r = dst + (size_t)(mBase + v) * D + l16;
        r[0]  = acc0[v];
        r[16] = acc1[v];
        r[32] = acc2[v];
        r[48] = acc3[v];
    }
}

// ---------------- phase 4: edge gather + mean-normalized scatter-add ------
// 16 threads per edge, float4 each (64 floats / edge). xr is L2-resident.
__global__ void __launch_bounds__(256)
rgcn_scatter(const int* __restrict__ srcArr,
             const int* __restrict__ dstArr,
             const int* __restrict__ etype,
             const int* __restrict__ counts,
             const float* __restrict__ xr,
             float* __restrict__ out) {
    const long long idx = (long long)blockIdx.x * blockDim.x + threadIdx.x;
    const long long e = idx >> 4;
    if (e >= N_EDGES) return;
    const int part = ((int)idx & 15) * 4;

    const int s = srcArr[e];
    const int d = dstArr[e];
    const int r = etype[e];
    // This edge exists, so counts[d*R + r] >= 1: no divide-by-zero branch.
    const float norm = 1.0f / (float)counts[d * N_REL + r];

    const float4 m = *(const float4*)(xr + ((size_t)r * N_NODES + s) * D + part);
    float* o = out + (size_t)d * D + part;
    atomicAdd(o + 0, m.x * norm);
    atomicAdd(o + 1, m.y * norm);
    atomicAdd(o + 2, m.z * norm);
    atomicAdd(o + 3, m.w * norm);
}

// ---------------- host launcher ----------------
extern "C" void kernel_launch(void* const* d_in, const int* in_sizes, int n_in,
                              void* d_out, int out_size, void* d_ws, size_t ws_size,
                              hipStream_t stream) {
    const float* x    = (const float*)d_in[0];   // [50000][64]
    const float* W    = (const float*)d_in[1];   // [6][64][64]
    const float* root = (const float*)d_in[2];   // [64][64]
    const float* bias = (const float*)d_in[3];   // [64]
    const int* eidx   = (const int*)d_in[4];     // [2][1600000]
    const int* etype  = (const int*)d_in[5];     // [1600000]
    float* out = (float*)d_out;

    const int* srcArr = eidx;
    const int* dstArr = eidx + N_EDGES;

    // Workspace layout: [counts: 300000 ints][xr: 6*50000*64 floats]
    int*   counts = (int*)d_ws;
    float* xr     = (float*)((char*)d_ws + (size_t)N_NODES * N_REL * sizeof(int));

    // 1) zero counters
    {
        int n = N_NODES * N_REL;
        rgcn_zero_counts<<<(n + 255) / 256, 256, 0, stream>>>(counts);
    }
    // 2) per-(dst, rel) in-degree
    rgcn_count_edges<<<(N_EDGES + 255) / 256, 256, 0, stream>>>(dstArr, etype, counts);
    // 3) WMMA GEMMs: xr[0..5] to workspace; out = x@root + bias
    {
        dim3 grid(GEMM_BLOCKS, N_REL + 1);
        rgcn_gemm_wmma<<<grid, 256, 0, stream>>>(x, W, root, bias, out, xr);
    }
    // 4) normalized scatter-add of edge messages into out
    {
        long long threads = (long long)N_EDGES * 16;
        int blocks = (int)((threads + 255) / 256);
        rgcn_scatter<<<blocks, 256, 0, stream>>>(srcArr, dstArr, etype, counts, xr, out);
    }
}